// EdgePreserve_63720134803593
// MI455X (gfx1250) — compile-verified
//
#include <hip/hip_runtime.h>
#include <stdint.h>

#define W_IMG 1024
#define H_IMG 1024
#define TILE_W 256
#define TILE_H 16
#define HALO_W (TILE_W + 2)   // 258
#define HALO_H (TILE_H + 2)   // 18
#define PITCH 264             // LDS row pitch (floats)

typedef __attribute__((ext_vector_type(4))) unsigned int u32x4;
typedef __attribute__((ext_vector_type(8))) int i32x8;
typedef __attribute__((ext_vector_type(4))) int i32x4;

#if defined(__has_builtin)
#  if __has_builtin(__builtin_amdgcn_tensor_load_to_lds)
#    define HAVE_TDM 1
#  endif
#endif

// sigma_color = 2 -> 1/(2*sc^2) = 0.125 ; fold log2(e) so we use native exp2
#define K2  0.18033688011112043f    // 0.125 * log2(e)
#define WS1 0.9998000199986667f     // exp(-1/(2*50^2))
#define WS2 0.9996000799893344f     // exp(-2/(2*50^2))

__device__ __forceinline__ void bil(float nb, float x0, float ws,
                                    float& num, float& den) {
  float d = nb - x0;
  float w = __builtin_amdgcn_exp2f(-K2 * d * d) * ws;
  num = fmaf(w, nb, num);
  den += w;
}

__global__ __launch_bounds__(256) void bilateral_pool_kernel(
    const float* __restrict__ x, float* __restrict__ out) {
  __shared__ float tile[HALO_H * PITCH];

  const int tid = threadIdx.x;
  const int oxt = blockIdx.x;   // 0..3  (256-wide column strip)
  const int oy  = blockIdx.y;   // 0..63 (output row == 16-tall input strip)
  const int img = blockIdx.z;   // 0..15

  const int gx0 = oxt * TILE_W - 1;
  const int gy0 = oy  * TILE_H - 1;

  const int rx0 = gx0 > 0 ? gx0 : 0;
  const int ry0 = gy0 > 0 ? gy0 : 0;
  const int rx1 = (gx0 + HALO_W) < W_IMG ? (gx0 + HALO_W) : W_IMG;
  const int ry1 = (gy0 + HALO_H) < H_IMG ? (gy0 + HALO_H) : H_IMG;
  const int w = rx1 - rx0;
  const int h = ry1 - ry0;
  const int col_off = rx0 - gx0;   // 0 or 1
  const int row_off = ry0 - gy0;   // 0 or 1

  const float* src = x + (size_t)img * (W_IMG * H_IMG);

#if defined(HAVE_TDM)
  if (tid == 0) {
    unsigned long long ga =
        (unsigned long long)(uintptr_t)(src + (size_t)ry0 * W_IMG + rx0);
    unsigned lds_byte = (unsigned)(row_off * PITCH + col_off) * 4u;

    // ---- D# group 0: count / lds_addr / global_addr / type=2 ----
    u32x4 g0 = {
      1u,                                          // count=1, user descriptor
      lds_byte,                                    // LDS byte address (tile @ 0)
      (unsigned)(ga & 0xFFFFFFFFull),
      (unsigned)((ga >> 32) & 0x01FFFFFFull) | (2u << 30)
    };

    // ---- D# group 1 ----
    // d0: workgroup_mask=0, data_size=2 (4B), iterate_enable=1
    // tensor_dim0 = tensor_dim1 = 1M (never OOB; region pre-clamped)
    i32x8 g1 = {
      (int)((2u << 16) | (1u << 19)),
      0,                                           // tensor_dim0[15:0] << 16 = 0
      (int)0x00000010u,                            // tensor_dim0 hi=0x10, dim1 lo=0
      (int)(0x10u | ((unsigned)w << 16)),          // tensor_dim1 hi=0x10, tile_dim0=w
      1,                                           // tile_dim1=1 row/iter, tile_dim2=0
      W_IMG,                                       // tensor_dim0_stride
      0, 0
    };

    // ---- D# group 2 (iterate mode): lds/global increments + count ----
    i32x4 g2 = {
      0,                                           // tensor_dim2 (unused w/ iterate)
      PITCH,                                       // lds_addr_increment (elements)
      W_IMG,                                       // global_addr_increment (elements)
      (int)(((unsigned)(h - 1)) << 16)             // iterate_count = h-1 -> h rows
    };
    i32x4 g3 = {0, 0, 0, 0};
    i32x8 g4 = {0, 0, 0, 0, 0, 0, 0, 0};          // extra arg on this toolchain

    __builtin_amdgcn_tensor_load_to_lds(g0, g1, g2, g3, g4, 0);
    __builtin_amdgcn_s_wait_tensorcnt(0);
  }
#else
  for (int idx = tid; idx < HALO_H * HALO_W; idx += 256) {
    int r = idx / HALO_W, cc = idx % HALO_W;
    int gy = gy0 + r, gx = gx0 + cc;
    if (gy >= 0 && gy < H_IMG && gx >= 0 && gx < W_IMG)
      tile[r * PITCH + cc] = src[(size_t)gy * W_IMG + gx];
  }
#endif
  __syncthreads();

  // ---- reflect-pad fix-up at image borders (block-uniform branches) ----
  if (gx0 < 0 && tid < HALO_H)
    tile[tid * PITCH + 0]   = tile[tid * PITCH + 2];
  if (gx0 + HALO_W > W_IMG && tid < HALO_H)
    tile[tid * PITCH + 257] = tile[tid * PITCH + 255];
  __syncthreads();
  if (gy0 < 0)
    for (int cc = tid; cc < HALO_W; cc += 256)
      tile[cc] = tile[2 * PITCH + cc];
  if (gy0 + HALO_H > H_IMG)
    for (int cc = tid; cc < HALO_W; cc += 256)
      tile[17 * PITCH + cc] = tile[15 * PITCH + cc];
  __syncthreads();

  // ---- bilateral + vertical pooling: sliding 3x3 register window ----
  const int c = tid + 1;
  float a0 = tile[0 * PITCH + c - 1], a1 = tile[0 * PITCH + c], a2 = tile[0 * PITCH + c + 1];
  float b0 = tile[1 * PITCH + c - 1], b1 = tile[1 * PITCH + c], b2 = tile[1 * PITCH + c + 1];
  float acc = 0.0f;
#pragma unroll
  for (int r = 1; r <= TILE_H; ++r) {
    float e0 = tile[(r + 1) * PITCH + c - 1];
    float e1 = tile[(r + 1) * PITCH + c];
    float e2 = tile[(r + 1) * PITCH + c + 1];
    float x0 = b1;
    float num = x0, den = 1.0f;      // center: w == 1 exactly
    bil(a0, x0, WS2, num, den); bil(a1, x0, WS1, num, den); bil(a2, x0, WS2, num, den);
    bil(b0, x0, WS1, num, den);                             bil(b2, x0, WS1, num, den);
    bil(e0, x0, WS2, num, den); bil(e1, x0, WS1, num, den); bil(e2, x0, WS2, num, den);
    acc += num * __builtin_amdgcn_rcpf(den);
    a0 = b0; a1 = b1; a2 = b2;
    b0 = e0; b1 = e1; b2 = e2;
  }

  // ---- horizontal 16-wide pool inside each wave32 (two groups of 16) ----
  acc += __shfl_xor(acc, 1, 32);
  acc += __shfl_xor(acc, 2, 32);
  acc += __shfl_xor(acc, 4, 32);
  acc += __shfl_xor(acc, 8, 32);
  if ((tid & 15) == 0) {
    int ox = oxt * 16 + (tid >> 4);
    out[((size_t)img * 64 + oy) * 64 + ox] = acc * (1.0f / 256.0f);
  }
}

extern "C" void kernel_launch(void* const* d_in, const int* in_sizes, int n_in,
                              void* d_out, int out_size, void* d_ws, size_t ws_size,
                              hipStream_t stream) {
  (void)in_sizes; (void)n_in; (void)out_size; (void)d_ws; (void)ws_size;
  const float* x = (const float*)d_in[0];
  float* out = (float*)d_out;
  dim3 grid(W_IMG / TILE_W, H_IMG / TILE_H, 16);   // (4, 64, 16)
  dim3 block(256);
  bilateral_pool_kernel<<<grid, block, 0, stream>>>(x, out);
}